// MRCNER_9620726743778
// MI455X (gfx1250) — compile-verified
//
#include <hip/hip_runtime.h>
#include <math.h>

typedef __attribute__((ext_vector_type(2))) float v2f;
typedef __attribute__((ext_vector_type(8))) float v8f;

#define H     768
#define H2    1536
#define S     256
#define BATCH 2
#define ROWS  (BATCH * S)   // 512 flattened (b,s) rows

// ---------------------------------------------------------------------------
// Kernel 1: row_proj / col_proj via f32 WMMA (16x16 tile per wave, K step 4).
//   blockIdx.y = 0 -> row_proj (w1 rows [0,768)),  1 -> col_proj (w1 rows [768,1536))
//   4 waves per 128-thread block, each wave owns one 16x16 output tile.
// ---------------------------------------------------------------------------
__global__ void proj_gemm_wmma(const float* __restrict__ hidden,
                               const float* __restrict__ w1,
                               float* __restrict__ rowp,
                               float* __restrict__ colp) {
    const int lane = threadIdx.x & 31;
    const int wave = threadIdx.x >> 5;
    const int tile = blockIdx.x * 4 + wave;          // 0 .. 3071
    const int tileM = (tile / (H2 / 16)) * 16;       // 0 .. 496
    const int tileN = (tile % (H2 / 16)) * 16;       // 0 .. 1520

    const float* __restrict__ Bw = w1 + (blockIdx.y ? (size_t)H * H2 : 0);
    float* __restrict__ Out = blockIdx.y ? colp : rowp;

    const int l16   = lane & 15;
    const int khalf = (lane >> 4) * 2;               // 0 (lanes 0-15) or 2 (lanes 16-31)
    const float* __restrict__ Arow = hidden + (size_t)(tileM + l16) * H;

    v8f c = {};
    for (int k0 = 0; k0 < H; k0 += 4) {
        // A 16x4 tile: lane L holds A[M=L%16][K = khalf + {0,1}]
        v2f a = *(const v2f*)(Arow + k0 + khalf);
        // B 4x16 tile: lane L holds B[K = khalf + {0,1}][N = L%16]
        v2f b;
        b.x = Bw[(size_t)(k0 + khalf + 0) * H2 + tileN + l16];
        b.y = Bw[(size_t)(k0 + khalf + 1) * H2 + tileN + l16];
        c = __builtin_amdgcn_wmma_f32_16x16x4_f32(false, a, false, b,
                                                  (short)0, c, false, false);
    }

    // C/D: VGPR v -> M = v + (lane>=16 ? 8 : 0), N = lane%16
    const int mbase = tileM + ((lane >> 4) * 8);
    const int n     = tileN + l16;
#pragma unroll
    for (int v = 0; v < 8; ++v)
        Out[(size_t)(mbase + v) * H2 + n] = c[v];
}

// ---------------------------------------------------------------------------
// Kernel 2: start/end logits. One wave per (which, row) GEMV over H=768.
//   1024 tasks total; 8 waves per block -> 128 blocks.
// ---------------------------------------------------------------------------
__global__ void span_logits(const float* __restrict__ hidden,
                            const float* __restrict__ w_start,
                            const float* __restrict__ b_start,
                            const float* __restrict__ w_end,
                            const float* __restrict__ b_end,
                            float* __restrict__ out) {
    const int lane = threadIdx.x & 31;
    const int gid  = blockIdx.x * 8 + (threadIdx.x >> 5); // 0 .. 1023
    const int which = gid >> 9;                           // 0 = start, 1 = end
    const int row   = gid & (ROWS - 1);

    const float* __restrict__ w = which ? w_end : w_start;
    const float  bias           = which ? b_end[0] : b_start[0];
    const float* __restrict__ h = hidden + (size_t)row * H;

    float acc = 0.0f;
    for (int k = lane; k < H; k += 32)
        acc = fmaf(h[k], w[k], acc);
#pragma unroll
    for (int off = 16; off > 0; off >>= 1)
        acc += __shfl_down(acc, off, 32);
    if (lane == 0)
        out[which * ROWS + row] = acc + bias;
}

// ---------------------------------------------------------------------------
// Kernel 3: pairwise match logits.
//   block = 256 threads handles a 16(i) x 16(j) tile for one batch b.
//   K streamed in 128-wide chunks through LDS (pitch 129 -> conflict-free).
//   match[b,i,j] = sum_k gelu(row[i,k] + col[j,k] + b1[k]) * w2[k] + b2
// ---------------------------------------------------------------------------
#define KC 128

__global__ void pairwise_match(const float* __restrict__ rowp,
                               const float* __restrict__ colp,
                               const float* __restrict__ b1,
                               const float* __restrict__ w2,
                               const float* __restrict__ b2,
                               float* __restrict__ match) {
    __shared__ float rs[16][KC + 1];
    __shared__ float cs[16][KC + 1];
    __shared__ float w2s[KC];
    __shared__ float b1s[KC];

    const int t  = threadIdx.x;
    const int tx = t & 15;          // j within tile
    const int ty = t >> 4;          // i within tile
    const int j0 = blockIdx.x * 16;
    const int i0 = blockIdx.y * 16;
    const int bz = blockIdx.z;

    const size_t rbase = (size_t)(bz * S + i0) * H2;  // row_proj rows for this i-tile
    const size_t cbase = (size_t)(bz * S + j0) * H2;  // col_proj rows for this j-tile

    float acc = 0.0f;
    for (int kc = 0; kc < H2; kc += KC) {
#pragma unroll
        for (int p = 0; p < (16 * KC) / 256; ++p) {   // 8 elems per thread
            int idx = p * 256 + t;
            int r = idx >> 7;                         // KC == 128
            int c = idx & (KC - 1);
            rs[r][c] = rowp[rbase + (size_t)r * H2 + kc + c];
            cs[r][c] = colp[cbase + (size_t)r * H2 + kc + c];
        }
        if (t < KC) {
            w2s[t] = w2[kc + t];
            b1s[t] = b1[kc + t];
        }
        __syncthreads();

#pragma unroll 4
        for (int c = 0; c < KC; ++c) {
            float x = rs[ty][c] + cs[tx][c] + b1s[c];
            float g = 0.5f * x * (1.0f + erff(x * 0.70710678118654752f));
            acc = fmaf(g, w2s[c], acc);
        }
        __syncthreads();
    }

    const size_t oidx = ((size_t)(bz * S + i0 + ty) * S) + (j0 + tx);
    match[oidx] = acc + b2[0];
}

// ---------------------------------------------------------------------------
// Launch: inputs in setup_inputs() order:
//   0 hidden [2,256,768] f32      1 w_start [768,1]   2 b_start [1]
//   3 w_end [768,1]               4 b_end [1]
//   5 w1 [1536,1536]              6 b1 [1536]
//   7 w2 [1536,1]                 8 b2 [1]
// d_out = start(512) | end(512) | match(131072), all f32.
// ---------------------------------------------------------------------------
extern "C" void kernel_launch(void* const* d_in, const int* in_sizes, int n_in,
                              void* d_out, int out_size, void* d_ws, size_t ws_size,
                              hipStream_t stream) {
    const float* hidden  = (const float*)d_in[0];
    const float* w_start = (const float*)d_in[1];
    const float* b_start = (const float*)d_in[2];
    const float* w_end   = (const float*)d_in[3];
    const float* b_end   = (const float*)d_in[4];
    const float* w1      = (const float*)d_in[5];
    const float* b1      = (const float*)d_in[6];
    const float* w2      = (const float*)d_in[7];
    const float* b2      = (const float*)d_in[8];
    float* out = (float*)d_out;

    float* rowp = (float*)d_ws;                       // [512,1536]
    float* colp = rowp + (size_t)ROWS * H2;           // [512,1536]

    // 1) projections via WMMA: 3072 tiles per projection, 4 waves/block
    proj_gemm_wmma<<<dim3(768, 2), 128, 0, stream>>>(hidden, w1, rowp, colp);

    // 2) start/end logits (independent of kernel 1)
    span_logits<<<128, 256, 0, stream>>>(hidden, w_start, b_start, w_end, b_end, out);

    // 3) pairwise GELU + dot with w2
    pairwise_match<<<dim3(S / 16, S / 16, BATCH), 256, 0, stream>>>(
        rowp, colp, b1, w2, b2, out + 2 * ROWS);
}